// unit_gcn_2714419331727
// MI455X (gfx1250) — compile-verified
//
#include <hip/hip_runtime.h>
#include <hip/hip_bf16.h>

typedef float v2f __attribute__((ext_vector_type(2)));
typedef float v8f __attribute__((ext_vector_type(8)));

// Problem dimensions (fixed by the reference)
#define NB   64      // batch N
#define CIN  64      // in channels
#define COUT 64      // out channels
#define KS   3       // num_subset
#define DCH  (COUT*KS)   // 192 projected channels
#define TT   128
#define VV   25
#define TV   (TT*VV)     // 3200 (divisible by 64)
#define GRP  8

// Workspace layout (float offsets)
#define OFF_Y    0ull
#define SZ_Y     ((size_t)NB*DCH*TV)          // 39,321,600
#define OFF_Z    (OFF_Y + SZ_Y)
#define SZ_Z     ((size_t)NB*COUT*TV)         // 13,107,200
#define OFF_NA   (OFF_Z + SZ_Z)
#define SZ_NA    ((size_t)KS*GRP*VV*VV)       // 15,000
#define OFF_P0   (OFF_NA + SZ_NA)
#define SZ_P0    ((size_t)DCH*8*2)
#define OFF_SS0  (OFF_P0 + SZ_P0)
#define SZ_SS0   ((size_t)DCH*2)
#define OFF_P1   (OFF_SS0 + SZ_SS0)
#define SZ_P1    ((size_t)COUT*8*2)
#define OFF_SS1  (OFF_P1 + SZ_P1)

// ---------------------------------------------------------------------------
// Kernel 1: y[n,d,t,v] = sum_c x0[n,c,t,v] * W[c,d] + bias[d]
// WMMA f32 16x16x4. One wave computes a 32(d) x 64(tv) macro-tile:
// 2 A-operands x 4 B-operands -> 8 accumulators per K-step, so each step
// issues 12 per-lane dword loads for 8 WMMAs (1.5 loads/WMMA) and W is
// reused 4x more than a single-tile wave. All dims divide exactly ->
// EXEC stays all-ones (WMMA requirement), no masking.
// ---------------------------------------------------------------------------
__global__ __launch_bounds__(32) void gemm_proj_wmma(
    const float* __restrict__ x0, const float* __restrict__ W,
    const float* __restrict__ bias, float* __restrict__ y)
{
    const int lane = threadIdx.x;
    const int half = lane >> 4;      // 0: lanes 0-15, 1: lanes 16-31
    const int lm   = lane & 15;
    const int tv0  = blockIdx.x * 64;
    const int d0   = blockIdx.y * 32;
    const int n    = blockIdx.z;

    const float* xb = x0 + (size_t)n * CIN * TV + tv0;

    v8f acc[2][4] = {};
#pragma unroll
    for (int ks = 0; ks < CIN / 4; ++ks) {
        const int kb = ks * 4;
        // A (16x4, M=d, K=c): lanes 0-15 hold K=kb,kb+1; lanes 16-31 hold kb+2,kb+3
        const int ka = kb + (half ? 2 : 0);
        v2f a[2];
#pragma unroll
        for (int i = 0; i < 2; ++i) {
            a[i].x = W[(size_t)(ka    ) * DCH + d0 + i * 16 + lm];
            a[i].y = W[(size_t)(ka + 1) * DCH + d0 + i * 16 + lm];
        }
        // B (4x16, K=c, N=tv): vgpr0 rows K=kb(+half), vgpr1 rows K=kb+2(+half)
        v2f b[4];
#pragma unroll
        for (int j = 0; j < 4; ++j) {
            b[j].x = xb[(size_t)(kb + half    ) * TV + j * 16 + lm];
            b[j].y = xb[(size_t)(kb + 2 + half) * TV + j * 16 + lm];
        }
#pragma unroll
        for (int i = 0; i < 2; ++i)
#pragma unroll
            for (int j = 0; j < 4; ++j)
                acc[i][j] = __builtin_amdgcn_wmma_f32_16x16x4_f32(
                    false, a[i], false, b[j], (short)0, acc[i][j], false, false);
    }

#pragma unroll
    for (int i = 0; i < 2; ++i) {
        float* yo = y + ((size_t)n * DCH + d0 + i * 16) * TV + tv0 + lm;
#pragma unroll
        for (int r = 0; r < 8; ++r) {
            const int m = r + half * 8;          // C/D layout: M = r (+8 for hi half)
            const float bv = bias[d0 + i * 16 + m];
#pragma unroll
            for (int j = 0; j < 4; ++j)
                yo[(size_t)m * TV + j * 16] = acc[i][j][r] + bv;
        }
    }
}

// ---------------------------------------------------------------------------
// Kernel 2: deterministic per-channel partial sums (sum, sumsq).
// grid = (C, 8); block reduces 8 batches worth of one channel.
// ---------------------------------------------------------------------------
__global__ __launch_bounds__(256) void chan_stats(
    const float* __restrict__ buf, float* __restrict__ partials, int C)
{
    const int ch = blockIdx.x, part = blockIdx.y, tid = threadIdx.x;
    float s = 0.f, q = 0.f;
    for (int nn = part * 8; nn < part * 8 + 8; ++nn) {
        const float* p = buf + ((size_t)nn * C + ch) * TV;
        for (int i = tid; i < TV; i += 256) { float x = p[i]; s += x; q += x * x; }
    }
    __shared__ float ss[256], sq[256];
    ss[tid] = s; sq[tid] = q; __syncthreads();
    for (int o = 128; o > 0; o >>= 1) {
        if (tid < o) { ss[tid] += ss[tid + o]; sq[tid] += sq[tid + o]; }
        __syncthreads();
    }
    if (tid == 0) {
        partials[(size_t)(ch * 8 + part) * 2    ] = ss[0];
        partials[(size_t)(ch * 8 + part) * 2 + 1] = sq[0];
    }
}

// Kernel 3: fold partials -> per-channel affine (scale = g*rsqrt(var+eps),
// shift = beta - mean*scale). Deterministic fixed-order sum of 8 partials.
__global__ void bn_finalize(const float* __restrict__ partials,
                            const float* __restrict__ gamma,
                            const float* __restrict__ beta,
                            float* __restrict__ scale_shift, int C)
{
    const int ch = blockIdx.x * blockDim.x + threadIdx.x;
    if (ch >= C) return;
    float s = 0.f, q = 0.f;
    for (int p = 0; p < 8; ++p) {
        s += partials[(size_t)(ch * 8 + p) * 2];
        q += partials[(size_t)(ch * 8 + p) * 2 + 1];
    }
    const float cnt = (float)NB * (float)TV;
    const float m   = s / cnt;
    const float var = q / cnt - m * m;
    const float inv = rsqrtf(var + 1e-5f);
    const float sc  = gamma[ch] * inv;
    scale_shift[ch * 2]     = sc;
    scale_shift[ch * 2 + 1] = beta[ch] - m * sc;
}

// Kernel 4: norm_A[k,g,v,w] = DecoupleA[k,g,v,w] / (colsum_v + 0.001)
__global__ void norm_adj(const float* __restrict__ DA, float* __restrict__ normA)
{
    const int kg = blockIdx.x;          // 0..23 = k*8+g
    const int w  = threadIdx.x;
    if (w >= VV) return;
    const float* A = DA + (size_t)kg * VV * VV;
    float s = 0.f;
    for (int v = 0; v < VV; ++v) s += A[v * VV + w];
    const float inv = 1.0f / (s + 0.001f);
    float* o = normA + (size_t)kg * VV * VV;
    for (int v = 0; v < VV; ++v) o[v * VV + w] = A[v * VV + w] * inv;
}

// ---------------------------------------------------------------------------
// Kernel 5: graph matmul with fused BN0:
//   z[n,c,t,w] = sum_{k,v} (y[n,k*64+c,t,v]*sc0 + sh0) * normA[k, c&7, v, w]
// WMMA f32 16x16x4: M=t tile (8 waves x 16), N=w (two tiles, zero-padded),
// K = (k,v) -> 3 * 7 steps of 4 (v padded 25->28 with zeros on both operands).
// normA for this block's group staged in LDS, zero-padded to [3][28][32];
// each A operand feeds both w-tile accumulators.
// ---------------------------------------------------------------------------
__global__ __launch_bounds__(256) void graph_wmma(
    const float* __restrict__ y, const float* __restrict__ normA,
    const float* __restrict__ ss0, float* __restrict__ z)
{
    __shared__ float sB[KS][28][32];
    const int tid = threadIdx.x;
    const int nc  = blockIdx.x;          // n*64 + c
    const int n   = nc >> 6, c = nc & 63, g = c & 7;

    for (int i = tid; i < KS * 28 * 32; i += 256) {
        const int k = i / (28 * 32), r = i % (28 * 32), v = r >> 5, w = r & 31;
        sB[k][v][w] = (v < VV && w < VV)
                    ? normA[((size_t)(k * GRP + g) * VV + v) * VV + w] : 0.0f;
    }
    __syncthreads();

    const int wave = tid >> 5, lane = tid & 31;
    const int half = lane >> 4, lm = lane & 15;
    const int t0   = wave * 16;

    float sc[KS], sh[KS];
#pragma unroll
    for (int k = 0; k < KS; ++k) {
        sc[k] = ss0[(k * COUT + c) * 2];
        sh[k] = ss0[(k * COUT + c) * 2 + 1];
    }

    v8f acc0 = {}, acc1 = {};
#pragma unroll
    for (int k = 0; k < KS; ++k) {
        const float* yk = y + ((size_t)n * DCH + k * COUT + c) * TV
                            + (size_t)(t0 + lm) * VV;
#pragma unroll
        for (int j = 0; j < 7; ++j) {
            const int vb = j * 4;
            const int va = vb + (half ? 2 : 0);
            v2f a, b0, b1;
            a.x = (va     < VV) ? yk[va    ] * sc[k] + sh[k] : 0.0f;
            a.y = (va + 1 < VV) ? yk[va + 1] * sc[k] + sh[k] : 0.0f;
            b0.x = sB[k][vb + half    ][lm];
            b0.y = sB[k][vb + 2 + half][lm];
            b1.x = sB[k][vb + half    ][16 + lm];
            b1.y = sB[k][vb + 2 + half][16 + lm];
            acc0 = __builtin_amdgcn_wmma_f32_16x16x4_f32(false, a, false, b0,
                                                         (short)0, acc0, false, false);
            acc1 = __builtin_amdgcn_wmma_f32_16x16x4_f32(false, a, false, b1,
                                                         (short)0, acc1, false, false);
        }
    }

    float* zo = z + (size_t)nc * TV + (size_t)t0 * VV;
#pragma unroll
    for (int r = 0; r < 8; ++r) {
        const int m = r + half * 8;
        zo[(size_t)m * VV + lm] = acc0[r];
        if (lm < VV - 16) zo[(size_t)m * VV + 16 + lm] = acc1[r];
    }
}

// Kernel 6: out = relu(z*scale + shift + x0), channel c = (i/TV)%64
__global__ __launch_bounds__(256) void epilogue(
    const float* __restrict__ z, const float* __restrict__ x0,
    const float* __restrict__ ss, float* __restrict__ out, int total)
{
    const int i = blockIdx.x * blockDim.x + threadIdx.x;
    if (i >= total) return;
    const int c = (i / TV) & 63;
    const float v = z[i] * ss[c * 2] + ss[c * 2 + 1] + x0[i];
    out[i] = v > 0.0f ? v : 0.0f;
}

extern "C" void kernel_launch(void* const* d_in, const int* in_sizes, int n_in,
                              void* d_out, int out_size, void* d_ws, size_t ws_size,
                              hipStream_t stream) {
    const float* x0     = (const float*)d_in[0];
    const float* W      = (const float*)d_in[1];
    const float* bias   = (const float*)d_in[2];
    const float* DA     = (const float*)d_in[3];
    const float* g0     = (const float*)d_in[4];
    const float* b0     = (const float*)d_in[5];
    const float* g1     = (const float*)d_in[6];
    const float* b1     = (const float*)d_in[7];
    float* out = (float*)d_out;
    float* ws  = (float*)d_ws;

    float* y    = ws + OFF_Y;
    float* z    = ws + OFF_Z;
    float* nA   = ws + OFF_NA;
    float* p0   = ws + OFF_P0;
    float* ss0  = ws + OFF_SS0;
    float* p1   = ws + OFF_P1;
    float* ss1  = ws + OFF_SS1;

    // 1) projection GEMM (WMMA f32, 32x64 macro-tile per wave)
    gemm_proj_wmma<<<dim3(TV / 64, DCH / 32, NB), 32, 0, stream>>>(x0, W, bias, y);
    // 2) BN0 stats + finalize (deterministic two-pass)
    chan_stats<<<dim3(DCH, 8), 256, 0, stream>>>(y, p0, DCH);
    bn_finalize<<<1, DCH, 0, stream>>>(p0, g0, b0, ss0, DCH);
    // 3) normalized adjacency (3*8 distinct 25x25 blocks)
    norm_adj<<<KS * GRP, 32, 0, stream>>>(DA, nA);
    // 4) graph matmul with fused BN0 (WMMA f32)
    graph_wmma<<<NB * COUT, 256, 0, stream>>>(y, nA, ss0, z);
    // 5) BN stats + finalize
    chan_stats<<<dim3(COUT, 8), 256, 0, stream>>>(z, p1, COUT);
    bn_finalize<<<1, COUT, 0, stream>>>(p1, g1, b1, ss1, COUT);
    // 6) BN + residual + ReLU
    const int total = NB * COUT * TV;
    epilogue<<<(total + 255) / 256, 256, 0, stream>>>(z, x0, ss1, out, total);
}